// SpikeFP64ScaleBy2K_29875792511521
// MI455X (gfx1250) — compile-verified
//
#include <hip/hip_runtime.h>
#include <cstdint>

// SpikeFP64ScaleBy2K: each row is an FP64 value encoded as 64 float32 "bits"
// (element 0 = sign, 1..11 = exponent MSB-first, 12..63 = mantissa MSB-first).
// out = x with exponent += integer_value(k)  (mod 2^11), unless k == +-0 -> out = x.
// Pure streaming problem: 768 MB total traffic, ~33us floor @ 23.3 TB/s.
// wave32 strategy: 1 row per wave iteration, lane l owns elements l and l+32,
// pack bits with ballot_w32 (uniform -> SALU bit logic), bitreverse for the
// MSB-first fields, NT cache hints + global_prefetch for the streaming path.

#define ROWS_PER_WAVE 16
#define PF_DIST 8  // rows ahead to prefetch (8 * 256B = 2KB per stream)

__global__ __launch_bounds__(256) void spike_fp64_scale_kernel(
    const float* __restrict__ x, const float* __restrict__ k,
    float* __restrict__ out, int nrows)
{
    const int lane = threadIdx.x & 31;
    const long long wave = (long long)((blockIdx.x * blockDim.x + threadIdx.x) >> 5);
    const long long row0 = wave * ROWS_PER_WAVE;

#pragma unroll 4
    for (int r = 0; r < ROWS_PER_WAVE; ++r) {
        const long long row = row0 + r;
        if (row >= nrows) return;              // wave-uniform guard
        const long long off = row * 64;

        if (row + PF_DIST < nrows) {           // warm lines ahead (global_prefetch_b8)
            __builtin_prefetch(x + off + 64LL * PF_DIST + lane, 0, 0);
            __builtin_prefetch(k + off + 64LL * PF_DIST + lane, 0, 0);
        }

        // Coalesced 128B loads; data is single-touch -> non-temporal.
        const float x0 = __builtin_nontemporal_load(x + off + lane);
        const float x1 = __builtin_nontemporal_load(x + off + lane + 32);
        const float k0 = __builtin_nontemporal_load(k + off + lane);
        const float k1 = __builtin_nontemporal_load(k + off + lane + 32);

        // Pack row bits: word bit j == element j of the row.
        const uint32_t xlo = __builtin_amdgcn_ballot_w32(x0 != 0.0f);
        const uint32_t xhi = __builtin_amdgcn_ballot_w32(x1 != 0.0f);
        const uint32_t klo = __builtin_amdgcn_ballot_w32(k0 != 0.0f);
        const uint32_t khi = __builtin_amdgcn_ballot_w32(k1 != 0.0f);
        const uint64_t xw = ((uint64_t)xhi << 32) | xlo;
        const uint64_t kw = ((uint64_t)khi << 32) | klo;

        // Exponent fields are MSB-first in the word -> bit-reverse the 11-bit field.
        const uint32_t ex = __builtin_bitreverse32((uint32_t)(xw >> 1) & 0x7FFu) >> 21;
        const uint32_t ek = __builtin_bitreverse32((uint32_t)(kw >> 1) & 0x7FFu) >> 21;

        // 16-bit significand of k: implicit 1 + top 15 mantissa bits (MSB-first).
        const uint32_t sig =
            (__builtin_bitreverse32((uint32_t)(kw >> 12) & 0x7FFFu) >> 17) | 0x8000u;

        // Reference's 4-layer barrel shifter: shift right by 15 - ((ek+1025)&15).
        const uint32_t sh   = (ek + 1025u) & 15u;          // (ek - 1023) mod 16
        const uint32_t kabs = (sig >> (15u - sh)) & 0x7FFu;

        // 11-bit two's complement if k negative.
        const uint32_t kfin = ((uint32_t)kw & 1u) ? ((0u - kabs) & 0x7FFu) : kabs;

        // New exponent, placed back MSB-first into bits 1..11.
        const uint32_t enew = (ex + kfin) & 0x7FFu;
        const uint32_t wfld = __builtin_bitreverse32(enew) >> 21;
        const uint64_t scaled = (xw & ~0xFFEull) | ((uint64_t)wfld << 1);

        // k == +-0  <=>  bits 1..63 of k all zero  -> pass x through.
        const uint64_t ow = ((kw & ~1ull) == 0) ? xw : scaled;

        // Unpack to 0.0f / 1.0f; coalesced 128B NT stores.
        __builtin_nontemporal_store((float)((ow >> lane) & 1ull),        out + off + lane);
        __builtin_nontemporal_store((float)((ow >> (lane + 32)) & 1ull), out + off + lane + 32);
    }
}

extern "C" void kernel_launch(void* const* d_in, const int* in_sizes, int n_in,
                              void* d_out, int out_size, void* d_ws, size_t ws_size,
                              hipStream_t stream) {
    const float* x = (const float*)d_in[0];
    const float* k = (const float*)d_in[1];
    float* out = (float*)d_out;

    const int nrows = in_sizes[0] / 64;                    // 1,048,576
    const long long waves = (nrows + ROWS_PER_WAVE - 1) / ROWS_PER_WAVE;
    const long long threads = waves * 32;
    const int block = 256;
    const int grid = (int)((threads + block - 1) / block); // 8192 blocks

    spike_fp64_scale_kernel<<<grid, block, 0, stream>>>(x, k, out, nrows);
}